// NodeUpdateGRU_86706799771814
// MI455X (gfx1250) — compile-verified
//
#include <hip/hip_runtime.h>
#include <hip/hip_bf16.h>
#include <math.h>

#define IN_DIM 512
#define HID    512
#define NPAR   200000

// streaming reduction config
#define NB_RED  512   // blocks in phase 1
#define BLK_RED 256   // threads per block (8 waves)

typedef __attribute__((ext_vector_type(2))) float v2f;
typedef __attribute__((ext_vector_type(4))) float f4v;
typedef __attribute__((ext_vector_type(8))) float v8f;

// ---------------------------------------------------------------------------
// Phase 1: column partial sums of parent_states (200000 x 512, row-major).
// Thread t in a block owns float4-column-group cg = t%128 (columns 4cg..4cg+3)
// and row-offset ro = t/128; rows stride by 2*gridDim. Coalesced b128 loads.
// Block-level LDS combine -> one 512-float partial per block. Deterministic.
// ---------------------------------------------------------------------------
__global__ void colsum_kernel(const float* __restrict__ X,
                              float* __restrict__ partial) {
    __shared__ f4v sh[128];
    const int tid = threadIdx.x;
    const int cg  = tid & 127;
    const int ro  = tid >> 7;
    const int rstride = 2 * gridDim.x;

    const f4v* __restrict__ Xv = (const f4v*)X;   // row r -> Xv[r*128 + cg]
    f4v acc = {0.f, 0.f, 0.f, 0.f};

    for (int r = 2 * (int)blockIdx.x + ro; r < NPAR; r += rstride) {
        // hint the streaming pattern (global_prefetch_b8; speculative, OOB-safe)
        __builtin_prefetch(&Xv[(size_t)(r + rstride) * 128 + cg], 0, 1);
        f4v v = Xv[(size_t)r * 128 + cg];
        acc.x += v.x; acc.y += v.y; acc.z += v.z; acc.w += v.w;
    }

    if (tid >= 128) sh[cg] = acc;
    __syncthreads();
    if (tid < 128) {
        f4v o = sh[cg];
        acc.x += o.x; acc.y += o.y; acc.z += o.z; acc.w += o.w;
        f4v* P = (f4v*)(partial + (size_t)blockIdx.x * IN_DIM);
        P[cg] = acc;
    }
}

// ---------------------------------------------------------------------------
// Phase 2: reduce NB_RED partials per column, divide by NPAR -> agg (512).
// ---------------------------------------------------------------------------
__global__ void mean_kernel(const float* __restrict__ partial,
                            float* __restrict__ agg) {
    const int j = threadIdx.x;                    // 512 threads, 1 block
    float s = 0.f;
    for (int b = 0; b < NB_RED; ++b) s += partial[(size_t)b * IN_DIM + j];
    agg[j] = s * (1.0f / (float)NPAR);
}

// ---------------------------------------------------------------------------
// Phase 3: gi = W_ih @ agg + b_ih via V_WMMA_F32_16X16X4_F32 (fp32 WMMA).
// One wave (32 lanes) per 16-row tile; 96 tiles cover 3*HID = 1536 rows.
// A (16x4): lanes 0-15 hold rows M=0..15 with K={k,k+1}; lanes 16-31 K={k+2,k+3}.
// B (4x16): agg[k..k+3] broadcast across all 16 columns, same K indexing as A
// (columns identical => result independent of B's N-lane permutation).
// D column N=0 = GEMV result: lane 0 holds M=0..7 in c[0..7], lane 16 M=8..15.
// Two accumulators break the wmma D->C dependency chain.
// ---------------------------------------------------------------------------
__global__ void gemv_wmma_kernel(const float* __restrict__ W,
                                 const float* __restrict__ agg,
                                 const float* __restrict__ bias,
                                 float* __restrict__ gi) {
    const int lane  = threadIdx.x;          // 32 threads = 1 wave (wave32)
    const int m0    = blockIdx.x * 16;      // 96 blocks
    const int row   = lane & 15;
    const int khalf = (lane >> 4) << 1;     // 0 for lanes 0-15, 2 for 16-31

    const float* __restrict__ Wr = W + (size_t)(m0 + row) * IN_DIM;

    v8f c0 = {};
    v8f c1 = {};
    for (int k = 0; k < IN_DIM; k += 8) {
        const int kk0 = k + khalf;
        const int kk1 = k + 4 + khalf;
        v2f a0; a0.x = Wr[kk0];  a0.y = Wr[kk0 + 1];
        v2f b0; b0.x = agg[kk0]; b0.y = agg[kk0 + 1];
        v2f a1; a1.x = Wr[kk1];  a1.y = Wr[kk1 + 1];
        v2f b1; b1.x = agg[kk1]; b1.y = agg[kk1 + 1];
        c0 = __builtin_amdgcn_wmma_f32_16x16x4_f32(
                 false, a0, false, b0, (short)0, c0, false, false);
        c1 = __builtin_amdgcn_wmma_f32_16x16x4_f32(
                 false, a1, false, b1, (short)0, c1, false, false);
    }

    // extract column N=0 and add bias_ih
    if (lane == 0) {
#pragma unroll
        for (int i = 0; i < 8; ++i)
            gi[m0 + i] = c0[i] + c1[i] + bias[m0 + i];
    } else if (lane == 16) {
#pragma unroll
        for (int i = 0; i < 8; ++i)
            gi[m0 + 8 + i] = c0[i] + c1[i] + bias[m0 + 8 + i];
    }
}

// ---------------------------------------------------------------------------
// Phase 4: GRU gates with h = 0 (gh = bias_hh):
//   r = sigmoid(gi_r + bhh_r); z = sigmoid(gi_z + bhh_z);
//   n = tanh(gi_n + r * bhh_n); h_new = (1 - z) * n
// ---------------------------------------------------------------------------
__global__ void gru_gate_kernel(const float* __restrict__ gi,
                                const float* __restrict__ b_hh,
                                float* __restrict__ out) {
    const int j = threadIdx.x;               // 512 threads, 1 block
    const float gr = gi[j]           + b_hh[j];
    const float gz = gi[HID + j]     + b_hh[HID + j];
    const float gn = gi[2 * HID + j];
    const float hn = b_hh[2 * HID + j];
    const float r  = 1.f / (1.f + expf(-gr));
    const float z  = 1.f / (1.f + expf(-gz));
    const float n  = tanhf(gn + r * hn);
    out[j] = (1.f - z) * n;
}

// ---------------------------------------------------------------------------
extern "C" void kernel_launch(void* const* d_in, const int* in_sizes, int n_in,
                              void* d_out, int out_size, void* d_ws, size_t ws_size,
                              hipStream_t stream) {
    (void)in_sizes; (void)n_in; (void)out_size; (void)ws_size;

    const float* X    = (const float*)d_in[0];   // (200000, 512)
    const float* Wih  = (const float*)d_in[1];   // (1536, 512)
    const float* Whh  = (const float*)d_in[2];   // (1536, 512) — unused (h = 0)
    const float* bih  = (const float*)d_in[3];   // (1536,)
    const float* bhh  = (const float*)d_in[4];   // (1536,)
    (void)Whh;

    float* out = (float*)d_out;                  // (512,)
    float* ws  = (float*)d_ws;
    float* partial = ws;                               // NB_RED * 512 floats (1 MB)
    float* agg     = ws + (size_t)NB_RED * IN_DIM;     // 512 floats
    float* gi      = agg + IN_DIM;                     // 1536 floats

    colsum_kernel<<<NB_RED, BLK_RED, 0, stream>>>(X, partial);
    mean_kernel<<<1, IN_DIM, 0, stream>>>(partial, agg);
    gemv_wmma_kernel<<<(3 * HID) / 16, 32, 0, stream>>>(Wih, agg, bih, gi);
    gru_gate_kernel<<<1, HID, 0, stream>>>(gi, bhh, out);
}